// GAUSConv_17600775979448
// MI455X (gfx1250) — compile-verified
//
#include <hip/hip_runtime.h>

typedef __attribute__((ext_vector_type(2))) float v2f;
typedef __attribute__((ext_vector_type(8))) float v8f;

#define N_NODES 50000
#define N_EDGES 800000
#define D_IN    128
#define KK      4
#define FF      32
#define KF      128   // KK * FF

// ---------- order-preserving float<->uint mapping (for atomicMax on floats) ----
__device__ __forceinline__ unsigned ordOf(float f) {
  unsigned u = __float_as_uint(f);
  return (u >> 31) ? ~u : (u | 0x80000000u);
}
__device__ __forceinline__ float ofOrd(unsigned o) {
  return (o >> 31) ? __uint_as_float(o & 0x7fffffffu) : __uint_as_float(~o);
}

// ---------- init: out = bias (broadcast), nodemax = ord(-inf)=0, nodesum = 0 --
__global__ __launch_bounds__(256) void init_ws(const float* __restrict__ bias,
                                               float* __restrict__ out,
                                               unsigned* __restrict__ nodemax,
                                               float* __restrict__ nodesum) {
  int t = blockIdx.x * 256 + threadIdx.x;
  if (t < N_NODES * KF) out[t] = bias[t & (KF - 1)];
  if (t < N_NODES * KK) { nodemax[t] = 0u; nodesum[t] = 0.0f; }
}

// ---------- kernel 1: ft[N,128] = feat[N,128] @ W^T[128,128] via fp32 WMMA ----
// Block = 256 thr = 8 waves. Block owns 16 output rows (50000 = 3125*16 exact),
// wave w owns 16 output cols. K iterated in steps of 4 -> 32 chained WMMAs.
// A tile (16x128 f32, 8KB) staged via LDS and shared by all 8 waves.
__global__ __launch_bounds__(256) void fc_wmma(const float* __restrict__ feat,
                                               const float* __restrict__ W,
                                               float* __restrict__ ft) {
  __shared__ float As[16 * D_IN];
  const int lane = threadIdx.x & 31;
  const int wave = threadIdx.x >> 5;
  const int row0 = blockIdx.x * 16;

  // cooperative stage of A tile (2048 floats = 512 float4, 2 per thread)
  for (int i = threadIdx.x; i < 16 * D_IN / 4; i += 256) {
    ((float4*)As)[i] = ((const float4*)(feat + row0 * D_IN))[i];
  }
  __syncthreads();

  const int col0 = wave * 16;
  const int ln   = lane & 15;          // row of A frag / col of B frag
  const int kh   = (lane >> 4) << 1;   // K sub-offset: lanes 0-15 -> K=0,1; 16-31 -> K=2,3

  v8f c = {};
  #pragma unroll 8
  for (int k = 0; k < D_IN; k += 4) {
    // A 16x4 layout: lane ln holds (K=kh, K=kh+1) of row ln
    v2f a; a.x = As[ln * D_IN + k + kh];
           a.y = As[ln * D_IN + k + kh + 1];
    // B 4x16 layout: VGPR r, lanes 0-15 -> K=r, lanes 16-31 -> K=r+2, N=ln
    // B[kq][n] = W^T[kq][col0+n] = W[col0+n][kq]
    v2f b; b.x = W[(col0 + ln) * D_IN + k + kh];
           b.y = W[(col0 + ln) * D_IN + k + kh + 1];
    c = __builtin_amdgcn_wmma_f32_16x16x4_f32(false, a, false, b,
                                              (short)0, c, false, false);
  }

  // C/D 16x16 f32: VGPR r, lanes 0-15 -> M=r, lanes 16-31 -> M=r+8, N=ln
  const int mo = (lane >> 4) << 3;
  #pragma unroll
  for (int r = 0; r < 8; ++r) {
    ft[(row0 + r + mo) * KF + col0 + ln] = c[r];
  }
}

// ---------- kernel 2: per-(edge,k) logits + segment max -----------------------
__global__ __launch_bounds__(256) void edge_logits(const float* __restrict__ ft,
                                                   const float* __restrict__ mu,
                                                   const float* __restrict__ sigma,
                                                   const int* __restrict__ src,
                                                   const int* __restrict__ dst,
                                                   float* __restrict__ lbuf,
                                                   unsigned* __restrict__ nodemax) {
  int t = blockIdx.x * 256 + threadIdx.x;
  if (t >= N_EDGES * KK) return;
  const int k = t & 3;
  const int e = t >> 2;
  const int s = src[e];
  const int d = dst[e];
  const float4* ps = (const float4*)(ft + (size_t)s * KF + k * FF);
  const float4* pd = (const float4*)(ft + (size_t)d * KF + k * FF);
  const float4* pm = (const float4*)(mu + k * FF);
  float acc = 0.0f;
  #pragma unroll
  for (int i = 0; i < FF / 4; ++i) {
    float4 a = ps[i], b = pd[i], m = pm[i];
    float dx = a.x - b.x - m.x;
    float dy = a.y - b.y - m.y;
    float dz = a.z - b.z - m.z;
    float dw = a.w - b.w - m.w;
    acc += dx * dx + dy * dy + dz * dz + dw * dw;
  }
  float logit = -sqrtf(acc) * sigma[k];
  lbuf[t] = logit;                                  // t == e*4 + k
  atomicMax(nodemax + (size_t)d * KK + k, ordOf(logit));
}

// ---------- kernel 3: e = exp(logit - segmax), segment sum --------------------
__global__ __launch_bounds__(256) void edge_exp(const int* __restrict__ dst,
                                                const unsigned* __restrict__ nodemax,
                                                float* __restrict__ lbuf,
                                                float* __restrict__ nodesum) {
  int t = blockIdx.x * 256 + threadIdx.x;
  if (t >= N_EDGES * KK) return;
  const int k = t & 3;
  const int e = t >> 2;
  const int d = dst[e];
  float mx = ofOrd(nodemax[(size_t)d * KK + k]);
  float ev = expf(lbuf[t] - mx);
  lbuf[t] = ev;
  atomicAdd(nodesum + (size_t)d * KK + k, ev);
}

// ---------- kernel 4: out[dst] += ft[src] * (e / segsum) ----------------------
// One wave per edge; lane owns 4 contiguous floats of the 128-wide row.
__global__ __launch_bounds__(256) void edge_scatter(const float* __restrict__ ft,
                                                    const int* __restrict__ src,
                                                    const int* __restrict__ dst,
                                                    const float* __restrict__ lbuf,
                                                    const float* __restrict__ nodesum,
                                                    float* __restrict__ out) {
  const int wave = threadIdx.x >> 5;
  const int lane = threadIdx.x & 31;
  const long e = (long)blockIdx.x * 8 + wave;
  if (e >= N_EDGES) return;
  const int s = src[e];
  const int d = dst[e];
  const int idx = lane * 4;        // 0..124
  const int k   = lane >> 3;       // idx >> 5
  const float a = lbuf[e * 4 + k] / nodesum[(size_t)d * KK + k];
  float4 v = *(const float4*)(ft + (size_t)s * KF + idx);
  float* po = out + (size_t)d * KF + idx;
  atomicAdd(po + 0, v.x * a);
  atomicAdd(po + 1, v.y * a);
  atomicAdd(po + 2, v.z * a);
  atomicAdd(po + 3, v.w * a);
}

extern "C" void kernel_launch(void* const* d_in, const int* in_sizes, int n_in,
                              void* d_out, int out_size, void* d_ws, size_t ws_size,
                              hipStream_t stream) {
  const float* feat  = (const float*)d_in[0];   // [N, 128]
  const float* W     = (const float*)d_in[1];   // [128, 128]
  const float* bias  = (const float*)d_in[2];   // [1, 4, 32]
  const float* mu    = (const float*)d_in[3];   // [1, 4, 32]
  const float* sigma = (const float*)d_in[4];   // [4, 1]
  const int*   src   = (const int*)d_in[5];     // [E]
  const int*   dst   = (const int*)d_in[6];     // [E]
  float* out = (float*)d_out;                   // [N, 4, 32]

  char* ws = (char*)d_ws;
  float*    ft      = (float*)(ws);                         // 25,600,000 B
  float*    lbuf    = (float*)(ws + 25600000);              // 12,800,000 B
  unsigned* nodemax = (unsigned*)(ws + 38400000);           //    800,000 B
  float*    nodesum = (float*)(ws + 39200000);              //    800,000 B

  init_ws<<<(N_NODES * KF + 255) / 256, 256, 0, stream>>>(bias, out, nodemax, nodesum);
  fc_wmma<<<N_NODES / 16, 256, 0, stream>>>(feat, W, ft);
  edge_logits<<<(N_EDGES * KK + 255) / 256, 256, 0, stream>>>(ft, mu, sigma, src, dst,
                                                              lbuf, nodemax);
  edge_exp<<<(N_EDGES * KK + 255) / 256, 256, 0, stream>>>(dst, nodemax, lbuf, nodesum);
  edge_scatter<<<(N_EDGES + 7) / 8, 256, 0, stream>>>(ft, src, dst, lbuf, nodesum, out);
}